// BalanceCrossEntropyLossV2_79817672228983
// MI455X (gfx1250) — compile-verified
//
#include <hip/hip_runtime.h>
#include <stdint.h>

// ---------------- problem constants ----------------
#define NSAMP 16
#define HW (640 * 640)            // 409600 pixels per sample
#define NEG_RATIO_F 3.0f
#define EPS_F 1e-8f

#define L1BINS 4096               // key1 = bits >> 19   (sign|exp|mant[22:19]) — loss<=100 => key<2144
#define L3BINS 128                // key3 = bits & 127
#define CHUNK 1024                // elements staged per TDM op (4 KB per array)
#define BPS1 25                   // blocks per sample, pass 1
#define CPB1 16                   // chunks per block, pass 1 (25*16*1024 == HW)
#define BPS23 64                  // blocks per sample, passes 2/3

static_assert(BPS1 * CPB1 * CHUNK == HW, "chunking must tile the image exactly");

// ---------------- workspace layout (dwords) ----------------
#define OFF_HCNT1 0
#define OFF_HSUM1 (OFF_HCNT1 + NSAMP * L1BINS)
#define OFF_HCNT2 (OFF_HSUM1 + NSAMP * L1BINS)
#define OFF_HSUM2 (OFF_HCNT2 + NSAMP * L1BINS)
#define OFF_HCNT3 (OFF_HSUM2 + NSAMP * L1BINS)
#define OFF_HSUM3 (OFF_HCNT3 + NSAMP * L3BINS)
#define OFF_STATS (OFF_HSUM3 + NSAMP * L3BINS)   // per sample: pos_cnt, neg_cnt, pos_sum, pad
#define OFF_SEL1  (OFF_STATS + NSAMP * 4)        // per sample: bin, rem_k, sum_above(bits), pad
#define OFF_SEL2  (OFF_SEL1 + NSAMP * 4)
#define WS_DWORDS (OFF_SEL2 + NSAMP * 4)         // ~1.02 MB total

// ---------------- CDNA5 TDM availability ----------------
#if __has_builtin(__builtin_amdgcn_tensor_load_to_lds) && __has_builtin(__builtin_amdgcn_s_wait_tensorcnt)
#define USE_TDM 1
#else
#define USE_TDM 0
#endif

typedef unsigned int u32x4 __attribute__((ext_vector_type(4)));
typedef int s32x4 __attribute__((ext_vector_type(4)));
typedef int s32x8 __attribute__((ext_vector_type(8)));

#if USE_TDM
// Build a D# for a contiguous 1-D tile of `nelems` f32 and issue TENSOR_LOAD_TO_LDS.
// Group 0: [1:0]=count=1, [63:32]=lds_addr, [120:64]=global_addr, [127:126]=type=2.
// Group 1: data_size=2 (4B) at [17:16]; tensor_dim0 at [79:48]; tensor_dim1=1 at [111:80];
//          tile_dim0 at [127:112]; tile_dim1/2 = 0 (unused); dim0 stride at [207:160].
__device__ __forceinline__ void tdm_load_1d_f32(const float* gsrc, uint32_t lds_byte_off,
                                                uint32_t nelems) {
  uint64_t ga = (uint64_t)(uintptr_t)gsrc;
  u32x4 g0;
  g0.x = 1u;                                      // count=1, user descriptor
  g0.y = lds_byte_off;                            // lds_addr (bytes, wave-relative)
  g0.z = (uint32_t)ga;                            // global_addr[31:0]
  g0.w = (uint32_t)(ga >> 32) | (2u << 30);       // global_addr[56:32] | type=2
  s32x8 g1;
  g1[0] = (int)(2u << 16);                        // workgroup_mask=0, data_size=4B
  g1[1] = (int)((nelems & 0xFFFFu) << 16);        // tensor_dim0[15:0]
  g1[2] = (int)((nelems >> 16) | (1u << 16));     // tensor_dim0[31:16] | tensor_dim1=1
  g1[3] = (int)((nelems & 0xFFFFu) << 16);        // tensor_dim1 hi=0 | tile_dim0=nelems
  g1[4] = 0;                                      // tile_dim1=0, tile_dim2=0 (unused)
  g1[5] = (int)nelems;                            // tensor_dim0_stride lo
  g1[6] = 0;
  g1[7] = 0;
  s32x4 z4 = (s32x4)0;                            // groups 2/3 unused (<=2D)
#if defined(__clang_major__) && (__clang_major__ >= 23)
  s32x8 z8 = (s32x8)0;
  __builtin_amdgcn_tensor_load_to_lds(g0, g1, z4, z4, z8, 0);
#else
  __builtin_amdgcn_tensor_load_to_lds(g0, g1, z4, z4, 0);
#endif
}
#endif // USE_TDM

// ---------------- math helpers ----------------
__device__ __forceinline__ float bce_loss(float p, float g) {
  float lp = fmaxf(logf(p), -100.0f);
  float l1 = fmaxf(log1pf(-p), -100.0f);
  return -(g * lp + (1.0f - g) * l1);
}

// ---------------- kernel: zero workspace ----------------
__global__ void k_zero(uint32_t* __restrict__ ws, int ndw) {
  for (int i = blockIdx.x * blockDim.x + threadIdx.x; i < ndw; i += gridDim.x * blockDim.x)
    ws[i] = 0u;
}

// ---------------- pass 1: stats + level-1 count/sum histogram ----------------
__global__ __launch_bounds__(256) void k1_hist(const float* __restrict__ pred,
                                               const float* __restrict__ gt,
                                               const float* __restrict__ mask,
                                               uint32_t* __restrict__ ws) {
  __shared__ uint32_t hcnt[L1BINS];
  __shared__ float hsum[L1BINS];
  __shared__ float red[3][8];
#if USE_TDM
  __shared__ __align__(16) float sbuf[2][3][CHUNK];  // double-buffered TDM staging (24 KB)
#endif
  const int tid = threadIdx.x;
  const int sample = blockIdx.x / BPS1;
  const int blk = blockIdx.x % BPS1;
  for (int i = tid; i < L1BINS; i += 256) { hcnt[i] = 0u; hsum[i] = 0.0f; }

  const size_t sbase = (size_t)sample * HW;
  const float* srcs[3] = {pred + sbase, gt + sbase, mask + sbase};
  float accPos = 0.0f, accNeg = 0.0f, accPSum = 0.0f;

  auto proc1 = [&](float p, float g, float m) {
    float loss = bce_loss(p, g);
    float pos = g * m;
    float neg = (1.0f - g) * m;
    accPos += pos;
    accNeg += neg;
    accPSum += loss * pos;
    float nl = loss * neg;
    if (nl > 0.0f) {
      uint32_t b = __float_as_uint(nl) >> 19;   // monotonic key, < 4096 for loss <= 100
      atomicAdd(&hcnt[b], 1u);
      atomicAdd(&hsum[b], nl);
    }
  };

#if USE_TDM
  const uint32_t ldsBase = (uint32_t)(uintptr_t)&sbuf[0][0][0];
  const int c0 = blk * CPB1;
  if (tid < 32) {  // wave 0 drives the TDM (EXEC ignored by TDM; whole-wave branch skips it)
#pragma unroll
    for (int j = 0; j < 3; ++j)
      tdm_load_1d_f32(srcs[j] + (size_t)c0 * CHUNK, ldsBase + (uint32_t)((0 * 3 + j) * CHUNK) * 4u,
                      CHUNK);
  }
  for (int c = 0; c < CPB1; ++c) {
    const int buf = c & 1;
    if (tid < 32) {
      if (c + 1 < CPB1) {
#pragma unroll
        for (int j = 0; j < 3; ++j)
          tdm_load_1d_f32(srcs[j] + (size_t)(c0 + c + 1) * CHUNK,
                          ldsBase + (uint32_t)(((buf ^ 1) * 3 + j) * CHUNK) * 4u, CHUNK);
        __builtin_amdgcn_s_wait_tensorcnt(3);   // previous chunk's 3 ops retired (in-order)
      } else {
        __builtin_amdgcn_s_wait_tensorcnt(0);
      }
    }
    __syncthreads();                            // staged data visible to all waves
    float4 pv = ((const float4*)sbuf[buf][0])[tid];
    float4 gv = ((const float4*)sbuf[buf][1])[tid];
    float4 mv = ((const float4*)sbuf[buf][2])[tid];
    proc1(pv.x, gv.x, mv.x); proc1(pv.y, gv.y, mv.y);
    proc1(pv.z, gv.z, mv.z); proc1(pv.w, gv.w, mv.w);
    __syncthreads();                            // done reading buf before it is re-staged
  }
#else
  __syncthreads();
  const float4* P = (const float4*)srcs[0];
  const float4* G = (const float4*)srcs[1];
  const float4* M = (const float4*)srcs[2];
  const int nvec = HW / 4;
  for (int v = blk * 256 + tid; v < nvec; v += BPS1 * 256) {
    float4 pv = P[v], gv = G[v], mv = M[v];
    proc1(pv.x, gv.x, mv.x); proc1(pv.y, gv.y, mv.y);
    proc1(pv.z, gv.z, mv.z); proc1(pv.w, gv.w, mv.w);
  }
  __syncthreads();
#endif

  // wave32 reduce the 3 scalar stats, then block-combine and flush with atomics
  for (int off = 16; off; off >>= 1) {
    accPos += __shfl_down(accPos, off);
    accNeg += __shfl_down(accNeg, off);
    accPSum += __shfl_down(accPSum, off);
  }
  const int wid = tid >> 5, lane = tid & 31;
  if (lane == 0) { red[0][wid] = accPos; red[1][wid] = accNeg; red[2][wid] = accPSum; }
  __syncthreads();
  if (tid == 0) {
    float a = 0.0f, b = 0.0f, c = 0.0f;
    for (int w = 0; w < 8; ++w) { a += red[0][w]; b += red[1][w]; c += red[2][w]; }
    float* st = (float*)ws + OFF_STATS + sample * 4;
    atomicAdd(&st[0], a); atomicAdd(&st[1], b); atomicAdd(&st[2], c);
  }
  uint32_t* gC = ws + OFF_HCNT1 + sample * L1BINS;
  float* gS = (float*)ws + OFF_HSUM1 + sample * L1BINS;
  for (int i = tid; i < L1BINS; i += 256) {
    if (hcnt[i]) { atomicAdd(&gC[i], hcnt[i]); atomicAdd(&gS[i], hsum[i]); }
  }
}

// ---------------- passes 2 & 3: restricted refinement histograms ----------------
template <int LEVEL>
__global__ __launch_bounds__(256) void k23_hist(const float* __restrict__ pred,
                                                const float* __restrict__ gt,
                                                const float* __restrict__ mask,
                                                uint32_t* __restrict__ ws) {
  constexpr int BINS = (LEVEL == 2) ? L1BINS : L3BINS;
  __shared__ uint32_t hcnt[BINS];
  __shared__ float hsum[BINS];
  const int tid = threadIdx.x;
  const int sample = blockIdx.x / BPS23;
  const int blk = blockIdx.x % BPS23;
  const uint32_t b1 = ws[OFF_SEL1 + sample * 4 + 0];         // 0xFFFFFFFF sentinel => no match
  const uint32_t b2 = (LEVEL == 3) ? ws[OFF_SEL2 + sample * 4 + 0] : 0u;
  for (int i = tid; i < BINS; i += 256) { hcnt[i] = 0u; hsum[i] = 0.0f; }
  __syncthreads();

  const size_t sbase = (size_t)sample * HW;
  const float4* P = (const float4*)(pred + sbase);
  const float4* G = (const float4*)(gt + sbase);
  const float4* M = (const float4*)(mask + sbase);
  const int nvec = HW / 4;
  const int stride = BPS23 * 256;

  auto proc1 = [&](float p, float g, float m) {
    float loss = bce_loss(p, g);
    float nl = loss * ((1.0f - g) * m);
    if (nl > 0.0f) {
      uint32_t u = __float_as_uint(nl);
      if ((u >> 19) == b1) {
        if (LEVEL == 2) {
          uint32_t key = (u >> 7) & 4095u;
          atomicAdd(&hcnt[key], 1u); atomicAdd(&hsum[key], nl);
        } else if (((u >> 7) & 4095u) == b2) {
          uint32_t key = u & 127u;
          atomicAdd(&hcnt[key], 1u); atomicAdd(&hsum[key], nl);
        }
      }
    }
  };

  for (int v = blk * 256 + tid; v < nvec; v += stride) {
    if (v + stride < nvec) {
      __builtin_prefetch((const void*)(P + v + stride), 0, 0);
      __builtin_prefetch((const void*)(G + v + stride), 0, 0);
      __builtin_prefetch((const void*)(M + v + stride), 0, 0);
    }
    float4 pv = P[v], gv = G[v], mv = M[v];
    proc1(pv.x, gv.x, mv.x); proc1(pv.y, gv.y, mv.y);
    proc1(pv.z, gv.z, mv.z); proc1(pv.w, gv.w, mv.w);
  }
  __syncthreads();

  uint32_t* gC = ws + ((LEVEL == 2) ? OFF_HCNT2 : OFF_HCNT3) + sample * BINS;
  float* gS = (float*)ws + ((LEVEL == 2) ? OFF_HSUM2 : OFF_HSUM3) + sample * BINS;
  for (int i = tid; i < BINS; i += 256) {
    if (hcnt[i]) { atomicAdd(&gC[i], hcnt[i]); atomicAdd(&gS[i], hsum[i]); }
  }
}

// ---------------- descending scan over a 4096-bin count/sum histogram ----------------
// level==1: k from stats; level==2: k = rem from sel1, accumulate sum_above
__global__ __launch_bounds__(256) void k_select(uint32_t* __restrict__ ws, int level) {
  __shared__ uint32_t scnt[256];
  __shared__ float ssum[256];
  const int n = blockIdx.x;
  const int t = threadIdx.x;

  const uint32_t* hc;
  const float* hs;
  uint32_t k;
  float baseSum;
  uint32_t* sel;
  if (level == 1) {
    hc = ws + OFF_HCNT1 + n * L1BINS;
    hs = (const float*)ws + OFF_HSUM1 + n * L1BINS;
    const float* st = (const float*)ws + OFF_STATS + n * 4;
    float nm = fminf(st[1], NEG_RATIO_F * st[0]);
    k = (uint32_t)fmaxf(nm, 0.0f);              // trunc toward zero, matches astype(int32)
    baseSum = 0.0f;
    sel = ws + OFF_SEL1 + n * 4;
  } else {
    hc = ws + OFF_HCNT2 + n * L1BINS;
    hs = (const float*)ws + OFF_HSUM2 + n * L1BINS;
    k = ws[OFF_SEL1 + n * 4 + 1];
    baseSum = __uint_as_float(ws[OFF_SEL1 + n * 4 + 2]);
    sel = ws + OFF_SEL2 + n * 4;
  }

  // thread t owns 16 contiguous bins, scanned descending: bin = 4095 - (t*16 + j)
  uint32_t lc = 0;
  float ls = 0.0f;
#pragma unroll
  for (int j = 0; j < 16; ++j) {
    int bin = 4095 - (t * 16 + j);
    lc += hc[bin];
    ls += hs[bin];
  }
  scnt[t] = lc;
  ssum[t] = ls;
  __syncthreads();

  uint32_t pref = 0, total = 0;
  float prefS = 0.0f;
  for (int u = 0; u < 256; ++u) {
    uint32_t c = scnt[u];
    total += c;
    if (u < t) { pref += c; prefS += ssum[u]; }
  }
  const uint32_t kk = (k > total) ? total : k;
  if (kk == 0) {
    if (t == 0) {
      sel[0] = 0xFFFFFFFFu;                     // sentinel bin: matches nothing downstream
      sel[1] = 0u;
      sel[2] = __float_as_uint(baseSum);
    }
    return;
  }
  if (pref < kk && pref + lc >= kk) {           // exactly one crossing thread
    uint32_t cum = pref;
    float s = prefS;
    for (int j = 0; j < 16; ++j) {
      int bin = 4095 - (t * 16 + j);
      uint32_t c = hc[bin];
      if (cum + c >= kk) {
        sel[0] = (uint32_t)bin;
        sel[1] = kk - cum;                      // >=1 elements still needed from this bin
        sel[2] = __float_as_uint(baseSum + s);  // sum of everything strictly above this bin
        break;
      }
      cum += c;
      s += hs[bin];
    }
  }
}

// ---------------- final: 128-bin scan, per-sample loss, mean ----------------
__global__ __launch_bounds__(32) void k_final(const uint32_t* __restrict__ ws,
                                              float* __restrict__ out) {
  const int lane = threadIdx.x;
  float per = 0.0f;
  if (lane < NSAMP) {
    const int n = lane;
    const float* st = (const float*)ws + OFF_STATS + n * 4;
    const float posC = st[0], negC = st[1], posS = st[2];
    const float negMin = fminf(negC, NEG_RATIO_F * posC);
    float topk = __uint_as_float(ws[OFF_SEL2 + n * 4 + 2]);
    const uint32_t rem = ws[OFF_SEL2 + n * 4 + 1];
    if (rem > 0u) {
      const uint32_t b1 = ws[OFF_SEL1 + n * 4 + 0];
      const uint32_t b2 = ws[OFF_SEL2 + n * 4 + 0];
      const uint32_t* hc = ws + OFF_HCNT3 + n * L3BINS;
      const float* hs = (const float*)ws + OFF_HSUM3 + n * L3BINS;
      uint32_t cum = 0;
      float sAbove = 0.0f;
      for (int bin = L3BINS - 1; bin >= 0; --bin) {
        const uint32_t c = hc[bin];
        if (cum + c >= rem) {
          // every element in this bin shares bit pattern (b1<<19)|(b2<<7)|bin — exact ties
          const float v = __uint_as_float((b1 << 19) | (b2 << 7) | (uint32_t)bin);
          topk += sAbove + (float)(rem - cum) * v;
          break;
        }
        cum += c;
        sAbove += hs[bin];
        if (bin == 0) topk += sAbove;           // safety: never triggers if counts consistent
      }
    }
    per = (posS + topk) / (posC + negMin + EPS_F);
  }
  for (int off = 16; off; off >>= 1) per += __shfl_down(per, off);
  if (lane == 0) out[0] = per * (1.0f / (float)NSAMP);
}

// ---------------- host launcher ----------------
extern "C" void kernel_launch(void* const* d_in, const int* in_sizes, int n_in, void* d_out,
                              int out_size, void* d_ws, size_t ws_size, hipStream_t stream) {
  (void)in_sizes; (void)n_in; (void)out_size; (void)ws_size;
  const float* pred = (const float*)d_in[0];
  const float* gt = (const float*)d_in[1];
  const float* mask = (const float*)d_in[2];
  float* out = (float*)d_out;
  uint32_t* ws = (uint32_t*)d_ws;   // requires ~1.02 MB of scratch

  k_zero<<<520, 256, 0, stream>>>(ws, WS_DWORDS);
  k1_hist<<<NSAMP * BPS1, 256, 0, stream>>>(pred, gt, mask, ws);
  k_select<<<NSAMP, 256, 0, stream>>>(ws, 1);
  k23_hist<2><<<NSAMP * BPS23, 256, 0, stream>>>(pred, gt, mask, ws);
  k_select<<<NSAMP, 256, 0, stream>>>(ws, 2);
  k23_hist<3><<<NSAMP * BPS23, 256, 0, stream>>>(pred, gt, mask, ws);
  k_final<<<1, 32, 0, stream>>>(ws, out);
}